// MultiScaleAggregator_57389353009605
// MI455X (gfx1250) — compile-verified
//
#include <hip/hip_runtime.h>
#include <math.h>

#define B_ 8
#define S_ 4096
#define F_ 32
#define OUTF 640            // F * 4 windows * 5 components
#define CHUNK 1024
#define HALO 1024           // >= w_max - 1 = 999
#define REGION (CHUNK + HALO)   // 2048
#define NTHREADS 256
#define EPT (REGION / NTHREADS) // 8 elements per thread

__global__ __launch_bounds__(NTHREADS)
void multiscale_kernel(const float* __restrict__ x, float* __restrict__ out)
{
    __shared__ float  xl[REGION];
    __shared__ double cs1[REGION + 1];
    __shared__ double cs2[REGION + 1];
    __shared__ double ws1[NTHREADS];
    __shared__ double ws2[NTHREADS];

    const int c   = blockIdx.x;   // t-chunk 0..3
    const int f   = blockIdx.y;   // feature 0..31
    const int b   = blockIdx.z;   // batch 0..7
    const int tid = threadIdx.x;
    const int t0  = c * CHUNK;

    const float* src = x + ((size_t)b * S_) * F_ + f;   // series stride = F_ elements

    // ---- CDNA5 async gather: global -> LDS, per-lane addresses ----
    #pragma unroll
    for (int k = 0; k < EPT; ++k) {
        int i  = tid + k * NTHREADS;          // region index 0..2047
        int tg = t0 - HALO + i;               // global t (may be < 0 for chunk 0)
        int tc = tg < 0 ? 0 : tg;             // edge ('replicate') padding with x[b,0,f]
        const float* gp = src + (size_t)tc * F_;
        unsigned lds_off = (unsigned)(unsigned long long)(&xl[i]);  // low 32 bits = LDS byte offset
        asm volatile("global_load_async_to_lds_b32 %0, %1, off"
                     :: "v"(lds_off), "v"(gp) : "memory");
    }
#if __has_builtin(__builtin_amdgcn_s_wait_asynccnt)
    __builtin_amdgcn_s_wait_asynccnt(0);
#else
    asm volatile("s_wait_asynccnt 0" ::: "memory");
#endif
    __syncthreads();

    // ---- fp64 prefix sums of x and x^2 over the 2048-element region ----
    const int base = tid * EPT;               // contiguous 8 elements per thread
    double s1 = 0.0, s2 = 0.0;
    #pragma unroll
    for (int k = 0; k < EPT; ++k) {
        double v = (double)xl[base + k];
        s1 += v;
        s2 += v * v;
    }
    ws1[tid] = s1;
    ws2[tid] = s2;
    __syncthreads();

    // Hillis-Steele inclusive scan over the 256 per-thread totals
    #pragma unroll
    for (int off = 1; off < NTHREADS; off <<= 1) {
        double a1 = ws1[tid], a2 = ws2[tid];
        double b1 = 0.0, b2 = 0.0;
        if (tid >= off) { b1 = ws1[tid - off]; b2 = ws2[tid - off]; }
        __syncthreads();
        ws1[tid] = a1 + b1;
        ws2[tid] = a2 + b2;
        __syncthreads();
    }
    double o1 = (tid > 0) ? ws1[tid - 1] : 0.0;
    double o2 = (tid > 0) ? ws2[tid - 1] : 0.0;

    if (tid == 0) { cs1[0] = 0.0; cs2[0] = 0.0; }
    double r1 = o1, r2 = o2;
    #pragma unroll
    for (int k = 0; k < EPT; ++k) {
        double v = (double)xl[base + k];
        r1 += v;
        r2 += v * v;
        cs1[base + k + 1] = r1;
        cs2[base + k + 1] = r2;
    }
    __syncthreads();

    // ---- emit: mean, std, approx_max, trend, anomaly for 4 windows ----
    const int   wlist[4] = {10, 50, 200, 1000};
    const float invw[4]  = {0.1f, 0.02f, 0.005f, 0.001f};

    float* outb = out + (size_t)b * S_ * OUTF + f;

    #pragma unroll
    for (int j = 0; j < CHUNK / NTHREADS; ++j) {
        int t  = t0 + j * NTHREADS + tid;     // global time index
        int li = HALO + j * NTHREADS + tid;   // region index of element t
        float  xt = xl[li];
        double a1 = cs1[li + 1];
        double a2 = cs2[li + 1];
        float* orow = outb + (size_t)t * OUTF;

        #pragma unroll
        for (int wi = 0; wi < 4; ++wi) {
            const int   w   = wlist[wi];
            const float iw  = invw[wi];
            float Sw1  = (float)(a1 - cs1[li + 1 - w]);   // window sum of x
            float Sw2  = (float)(a2 - cs2[li + 1 - w]);   // window sum of x^2
            float mean = Sw1 * iw;
            float var  = Sw2 * iw - mean * mean;
            float sd   = sqrtf(fmaxf(var, 1e-8f));
            float amax = mean + 2.0f * sd;
            float trend = (t >= w) ? (xt - xl[li - w]) : 0.0f;
            float anom  = (fabsf(xt - mean) > 3.0f * sd) ? 1.0f : 0.0f;

            float* o = orow + wi * (F_ * 5);
            o[0]      = mean;
            o[F_]     = sd;
            o[2 * F_] = amax;
            o[3 * F_] = trend;
            o[4 * F_] = anom;
        }
    }
}

extern "C" void kernel_launch(void* const* d_in, const int* in_sizes, int n_in,
                              void* d_out, int out_size, void* d_ws, size_t ws_size,
                              hipStream_t stream) {
    (void)in_sizes; (void)n_in; (void)out_size; (void)d_ws; (void)ws_size;
    const float* x = (const float*)d_in[0];
    float* out = (float*)d_out;
    dim3 grid(S_ / CHUNK, F_, B_);   // (4 chunks, 32 features, 8 batches) = 1024 blocks
    multiscale_kernel<<<grid, NTHREADS, 0, stream>>>(x, out);
}